// Difference_attention_2671469658234
// MI455X (gfx1250) — compile-verified
//
#include <hip/hip_runtime.h>
#include <hip/hip_bf16.h>

typedef float v2f __attribute__((ext_vector_type(2)));
typedef float v8f __attribute__((ext_vector_type(8)));

#define NQ 512
#define NKV 2562
#define NKV_PAD 2624   // 41 * 64
#define CDIM 64
#define NCHUNK 41
#define CHUNK 64       // NKV_PAD / NCHUNK
#define NQT (NQ / 16)  // 32 query tiles
#define SCALE (1.0f / 64.0f)

// ---- workspace layout (float offsets) ----
#define OFF_QW   0L
#define OFF_KW   (OFF_QW   + (long)NQ * CDIM)        // 32768
#define OFF_VW   (OFF_KW   + (long)NKV_PAD * CDIM)
#define OFF_EQ   (OFF_VW   + (long)NKV_PAD * CDIM)
#define OFF_IEQ  (OFF_EQ   + (long)NQ * CDIM)
#define OFF_EK   (OFF_IEQ  + (long)NQ * CDIM)
#define OFF_IEK  (OFF_EK   + (long)NKV_PAD * CDIM)
#define OFF_WSUM (OFF_IEK  + (long)NKV_PAD * CDIM)
#define OFF_EQC  (OFF_WSUM + 16L)
#define OFF_IEQC (OFF_EQC  + (long)NQ * 3)
#define OFF_EKC  (OFF_IEQC + (long)NQ * 3)
#define OFF_IEKC (OFF_EKC  + (long)NKV * 3)
#define OFF_PO   (OFF_IEKC + (long)NKV * 3 + 16L)
#define OFF_PM   (OFF_PO   + (long)NQT * NCHUNK * 16 * CDIM)
#define OFF_PS   (OFF_PM   + (long)NQT * NCHUNK * 16)

static __device__ __forceinline__ v8f wmma_f32_4(v2f a, v2f b, v8f c) {
    return __builtin_amdgcn_wmma_f32_16x16x4_f32(false, a, false, b, (short)0, c,
                                                 false, false);
}

// ---------------------------------------------------------------------------
// Kernel 1a: q_ = q@Wq ; [k|v] = kv@Wkv (rows padded to NKV_PAD with zeros);
//            wsum[d] = sum_c Wdelta[d,c].  One wave per 16x16 WMMA tile.
// ---------------------------------------------------------------------------
__global__ void diffattn_prep(const float* __restrict__ q,
                              const float* __restrict__ kv,
                              const float* __restrict__ Wq,
                              const float* __restrict__ Wkv,
                              const float* __restrict__ Wdelta,
                              float* __restrict__ q_ws,
                              float* __restrict__ k_ws,
                              float* __restrict__ v_ws,
                              float* __restrict__ wsum_ws) {
    if (blockIdx.x == 0 && threadIdx.x < 3) {
        float s = 0.f;
        for (int c = 0; c < CDIM; ++c) s += Wdelta[threadIdx.x * CDIM + c];
        wsum_ws[threadIdx.x] = s;
    }

    const int KV_TILES = (NKV_PAD / 16) * (128 / 16);  // 164*8 = 1312
    const int Q_TILES = (NQ / 16) * (CDIM / 16);       // 32*4  = 128
    int wave = (blockIdx.x * blockDim.x + threadIdx.x) >> 5;
    int lane = threadIdx.x & 31;
    if (wave >= KV_TILES + Q_TILES) return;

    int half = lane >> 4;
    int lm = lane & 15;

    if (wave < KV_TILES) {
        int mt = wave >> 3, nt = wave & 7;
        int m0 = mt * 16, n0 = nt * 16;
        int rowA = m0 + lm;
        bool inb = rowA < NKV;
        v8f acc = {};
        for (int kk = 0; kk < CDIM; kk += 4) {
            int ka = kk + 2 * half;
            v2f a, b;
            a.x = inb ? kv[rowA * CDIM + ka] : 0.f;
            a.y = inb ? kv[rowA * CDIM + ka + 1] : 0.f;
            b.x = Wkv[ka * 128 + n0 + lm];
            b.y = Wkv[(ka + 1) * 128 + n0 + lm];
            acc = wmma_f32_4(a, b, acc);
        }
        int col = n0 + lm;
#pragma unroll
        for (int r = 0; r < 8; ++r) {
            int row = m0 + r + 8 * half;
            if (col < CDIM)
                k_ws[row * CDIM + col] = acc[r];
            else
                v_ws[row * CDIM + (col - CDIM)] = acc[r];
        }
    } else {
        int t = wave - KV_TILES;
        int mt = t >> 2, nt = t & 3;
        int m0 = mt * 16, n0 = nt * 16;
        int rowA = m0 + lm;
        v8f acc = {};
        for (int kk = 0; kk < CDIM; kk += 4) {
            int ka = kk + 2 * half;
            v2f a, b;
            a.x = q[rowA * CDIM + ka];
            a.y = q[rowA * CDIM + ka + 1];
            b.x = Wq[ka * CDIM + n0 + lm];
            b.y = Wq[(ka + 1) * CDIM + n0 + lm];
            acc = wmma_f32_4(a, b, acc);
        }
        int col = n0 + lm;
#pragma unroll
        for (int r = 0; r < 8; ++r) {
            int row = m0 + r + 8 * half;
            q_ws[row * CDIM + col] = acc[r];
        }
    }
}

// ---------------------------------------------------------------------------
// Kernel 1b: precompute exp(+x)/exp(-x) for q_, k, q_coord, kv_coord.
// exp(-|a-b|) == min(e^a * e^-b, e^-a * e^b): moves 84M transcendentals to
// ~400K, inner loop becomes full-rate VALU mul/min/add.
// ---------------------------------------------------------------------------
__global__ void diffattn_exp(const float* __restrict__ q_ws,
                             const float* __restrict__ k_ws,
                             const float* __restrict__ q_coord,
                             const float* __restrict__ kv_coord,
                             float* __restrict__ Eq, float* __restrict__ iEq,
                             float* __restrict__ Ek, float* __restrict__ iEk,
                             float* __restrict__ Eqc, float* __restrict__ iEqc,
                             float* __restrict__ Ekc, float* __restrict__ iEkc) {
    const int NE_Q = NQ * CDIM;
    const int NE_K = NKV_PAD * CDIM;
    const int NE_QC = NQ * 3;
    const int NE_KC = NKV * 3;
    int i = blockIdx.x * blockDim.x + threadIdx.x;
    if (i < NE_Q) {
        float x = q_ws[i];
        Eq[i] = __expf(x);
        iEq[i] = __expf(-x);
    } else if (i < NE_Q + NE_K) {
        int j = i - NE_Q;
        float x = k_ws[j];
        Ek[j] = __expf(x);
        iEk[j] = __expf(-x);
    } else if (i < NE_Q + NE_K + NE_QC) {
        int j = i - NE_Q - NE_K;
        float x = q_coord[j];
        Eqc[j] = __expf(x);
        iEqc[j] = __expf(-x);
    } else if (i < NE_Q + NE_K + NE_QC + NE_KC) {
        int j = i - NE_Q - NE_K - NE_QC;
        float x = kv_coord[j];
        Ekc[j] = __expf(x);
        iEkc[j] = __expf(-x);
    }
}

// ---------------------------------------------------------------------------
// Kernel 2: partial attention. Grid = (NCHUNK, NQT). Each block: 16 queries x
// 64 kv. Scores via min(EqiEk, iEqEk), chunk-local softmax, p@v via WMMA.
// Writes partial out / max / sum for split-softmax combine.
// ---------------------------------------------------------------------------
__global__ void __launch_bounds__(256)
diffattn_score(const float* __restrict__ Eq, const float* __restrict__ iEq,
               const float* __restrict__ Ek, const float* __restrict__ iEk,
               const float* __restrict__ Eqc, const float* __restrict__ iEqc,
               const float* __restrict__ Ekc, const float* __restrict__ iEkc,
               const float* __restrict__ wsum_ws,
               const float* __restrict__ v_ws,
               float* __restrict__ po, float* __restrict__ pm,
               float* __restrict__ ps) {
    __shared__ float sEq[16][CDIM + 1];
    __shared__ float sIEq[16][CDIM + 1];
    __shared__ float sP[16][CHUNK + 8];  // stride 72: rows hit distinct banks
    __shared__ float sEqc[16][4];
    __shared__ float sIEqc[16][4];
    __shared__ float sRed[16][17];
    __shared__ float sMax[16];
    __shared__ float sSum[16];
    __shared__ float sW[3];

    const int tid = threadIdx.x;
    const int ch = blockIdx.x;
    const int qt = blockIdx.y;
    const int q0 = qt * 16;
    const int j0 = ch * CHUNK;

    for (int i = tid; i < 16 * CDIM; i += 256) {
        int r = i / CDIM, c = i % CDIM;
        sEq[r][c] = Eq[(q0 + r) * CDIM + c];
        sIEq[r][c] = iEq[(q0 + r) * CDIM + c];
    }
    if (tid < 48) {
        int r = tid / 3, d = tid % 3;
        sEqc[r][d] = Eqc[(q0 + r) * 3 + d];
        sIEqc[r][d] = iEqc[(q0 + r) * 3 + d];
    }
    if (tid < 3) sW[tid] = wsum_ws[tid];
    __syncthreads();

    // ---- scores: 16 consecutive threads share one kv row ----
    for (int p = tid; p < 16 * CHUNK; p += 256) {
        int qi = p & 15;
        int jl = p >> 4;
        int jj = j0 + jl;
        float s;
        if (jj < NKV) {
            s = 0.f;
            const float4* e4 = reinterpret_cast<const float4*>(Ek + jj * CDIM);
            const float4* i4 = reinterpret_cast<const float4*>(iEk + jj * CDIM);
#pragma unroll 4
            for (int c4 = 0; c4 < CDIM / 4; ++c4) {
                float4 ek = e4[c4];
                float4 ik = i4[c4];
                int c = 4 * c4;
                s += fminf(sEq[qi][c + 0] * ik.x, sIEq[qi][c + 0] * ek.x);
                s += fminf(sEq[qi][c + 1] * ik.y, sIEq[qi][c + 1] * ek.y);
                s += fminf(sEq[qi][c + 2] * ik.z, sIEq[qi][c + 2] * ek.z);
                s += fminf(sEq[qi][c + 3] * ik.w, sIEq[qi][c + 3] * ek.w);
            }
#pragma unroll
            for (int d = 0; d < 3; ++d) {
                float ekc = Ekc[jj * 3 + d];
                float ikc = iEkc[jj * 3 + d];
                s += fminf(sEqc[qi][d] * ikc, sIEqc[qi][d] * ekc) * sW[d];
            }
            s *= SCALE;
        } else {
            s = -__builtin_inff();  // padding -> p = 0
        }
        sP[qi][jl] = s;
    }
    __syncthreads();

    // ---- chunk-local row max (16 threads per row) ----
    const int qi = tid & 15;
    const int g = tid >> 4;
    {
        float m = -__builtin_inff();
        for (int jl = g; jl < CHUNK; jl += 16) m = fmaxf(m, sP[qi][jl]);
        sRed[qi][g] = m;
    }
    __syncthreads();
    if (tid < 16) {
        float m = sRed[tid][0];
#pragma unroll
        for (int i = 1; i < 16; ++i) m = fmaxf(m, sRed[tid][i]);
        sMax[tid] = m;
    }
    __syncthreads();

    // ---- exp & local sum ----
    {
        float mr = sMax[qi];
        float psum = 0.f;
        for (int jl = g; jl < CHUNK; jl += 16) {
            float e = __expf(sP[qi][jl] - mr);
            sP[qi][jl] = e;
            psum += e;
        }
        sRed[qi][g] = psum;
    }
    __syncthreads();
    if (tid < 16) {
        float s = 0.f;
#pragma unroll
        for (int i = 0; i < 16; ++i) s += sRed[tid][i];
        sSum[tid] = s;
    }
    __syncthreads();

    // ---- WMMA: partial_o = p @ v_chunk (unnormalized) ----
    const int wave = tid >> 5;
    const int lane = tid & 31;
    const int half = lane >> 4;
    const int lm = lane & 15;
    const long base = (long)(qt * NCHUNK + ch) * 16;
    if (wave < 4) {
        int n0 = wave * 16;
        v8f acc = {};
        for (int kk = 0; kk < CHUNK; kk += 4) {
            int ka = kk + 2 * half;
            v2f a, b;
            a.x = sP[lm][ka];
            a.y = sP[lm][ka + 1];
            b.x = v_ws[(j0 + ka) * CDIM + n0 + lm];
            b.y = v_ws[(j0 + ka + 1) * CDIM + n0 + lm];
            acc = wmma_f32_4(a, b, acc);
        }
#pragma unroll
        for (int r = 0; r < 8; ++r) {
            int row = r + 8 * half;
            po[(base + row) * CDIM + n0 + lm] = acc[r];
        }
    }
    if (tid < 16) {
        pm[base + tid] = sMax[tid];
        ps[base + tid] = sSum[tid];
    }
}

// ---------------------------------------------------------------------------
// Kernel 3: combine partials (log-sum-exp merge) then out = o @ Wproj + b.
// One block per query tile.
// ---------------------------------------------------------------------------
__global__ void __launch_bounds__(256)
diffattn_combine(const float* __restrict__ po, const float* __restrict__ pm,
                 const float* __restrict__ ps,
                 const float* __restrict__ Wproj,
                 const float* __restrict__ bproj, float* __restrict__ out) {
    __shared__ float sO[16][CDIM + 1];
    __shared__ float cm[16][NCHUNK + 1];
    __shared__ float cw[16][NCHUNK + 1];
    __shared__ float sM[16];
    __shared__ float sInv[16];

    const int tid = threadIdx.x;
    const int qt = blockIdx.x;
    const long pbase = (long)qt * NCHUNK * 16;

    for (int i = tid; i < 16 * NCHUNK; i += 256) {
        int r = i & 15, c = i >> 4;
        cm[r][c] = pm[pbase + c * 16 + r];
        cw[r][c] = ps[pbase + c * 16 + r];
    }
    __syncthreads();
    if (tid < 16) {
        float M = cm[tid][0];
        for (int c = 1; c < NCHUNK; ++c) M = fmaxf(M, cm[tid][c]);
        float S = 0.f;
        for (int c = 0; c < NCHUNK; ++c) S += cw[tid][c] * __expf(cm[tid][c] - M);
        sM[tid] = M;
        sInv[tid] = 1.0f / S;
    }
    __syncthreads();
    for (int i = tid; i < 16 * NCHUNK; i += 256) {
        int r = i & 15, c = i >> 4;
        cw[r][c] = __expf(cm[r][c] - sM[r]) * sInv[r];
    }
    __syncthreads();

    for (int idx = tid; idx < 16 * CDIM; idx += 256) {
        int r = idx >> 6, n = idx & 63;
        float a = 0.f;
        for (int c = 0; c < NCHUNK; ++c)
            a += po[(pbase + c * 16 + r) * CDIM + n] * cw[r][c];
        sO[r][n] = a;
    }
    __syncthreads();

    // ---- WMMA: out = o @ Wproj + b ----
    const int wave = tid >> 5;
    const int lane = tid & 31;
    const int half = lane >> 4;
    const int lm = lane & 15;
    if (wave < 4) {
        int n0 = wave * 16;
        v8f acc = {};
        for (int kk = 0; kk < CDIM; kk += 4) {
            int ka = kk + 2 * half;
            v2f a, b;
            a.x = sO[lm][ka];
            a.y = sO[lm][ka + 1];
            b.x = Wproj[ka * CDIM + n0 + lm];
            b.y = Wproj[(ka + 1) * CDIM + n0 + lm];
            acc = wmma_f32_4(a, b, acc);
        }
        float bb = bproj[n0 + lm];
#pragma unroll
        for (int r = 0; r < 8; ++r) {
            int row = r + 8 * half;
            out[(qt * 16 + row) * CDIM + n0 + lm] = acc[r] + bb;
        }
    }
}

extern "C" void kernel_launch(void* const* d_in, const int* in_sizes, int n_in,
                              void* d_out, int out_size, void* d_ws, size_t ws_size,
                              hipStream_t stream) {
    const float* q        = (const float*)d_in[0];
    const float* q_coord  = (const float*)d_in[1];
    const float* kv       = (const float*)d_in[2];
    const float* kv_coord = (const float*)d_in[3];
    const float* Wq       = (const float*)d_in[4];
    const float* Wkv      = (const float*)d_in[5];
    const float* Wdelta   = (const float*)d_in[6];
    const float* Wproj    = (const float*)d_in[7];
    const float* bproj    = (const float*)d_in[8];
    float* out = (float*)d_out;
    float* ws  = (float*)d_ws;

    float* q_ws    = ws + OFF_QW;
    float* k_ws    = ws + OFF_KW;
    float* v_ws    = ws + OFF_VW;
    float* Eqp     = ws + OFF_EQ;
    float* iEqp    = ws + OFF_IEQ;
    float* Ekp     = ws + OFF_EK;
    float* iEkp    = ws + OFF_IEK;
    float* wsum_ws = ws + OFF_WSUM;
    float* Eqc     = ws + OFF_EQC;
    float* iEqc    = ws + OFF_IEQC;
    float* Ekc     = ws + OFF_EKC;
    float* iEkc    = ws + OFF_IEKC;
    float* po      = ws + OFF_PO;
    float* pm      = ws + OFF_PM;
    float* ps      = ws + OFF_PS;

    // 1312 kv tiles + 128 q tiles = 1440 waves; 8 waves/block -> 180 blocks
    diffattn_prep<<<180, 256, 0, stream>>>(q, kv, Wq, Wkv, Wdelta,
                                           q_ws, k_ws, v_ws, wsum_ws);

    int exp_total = NQ * CDIM + NKV_PAD * CDIM + NQ * 3 + NKV * 3;
    diffattn_exp<<<(exp_total + 255) / 256, 256, 0, stream>>>(
        q_ws, k_ws, q_coord, kv_coord, Eqp, iEqp, Ekp, iEkp, Eqc, iEqc, Ekc, iEkc);

    dim3 sgrid(NCHUNK, NQT);
    diffattn_score<<<sgrid, 256, 0, stream>>>(Eqp, iEqp, Ekp, iEkp, Eqc, iEqc,
                                              Ekc, iEkc, wsum_ws, v_ws, po, pm, ps);

    diffattn_combine<<<NQT, 256, 0, stream>>>(po, pm, ps, Wproj, bproj, out);
}